// FeedbackLoss_44770739093937
// MI455X (gfx1250) — compile-verified
//
#include <hip/hip_runtime.h>

// ---------------------------------------------------------------------------
// FeedbackLoss on MI455X (gfx1250)
// B=8, H=W=1024, NSEG=513 (id 0 = background)
// Memory-bound: 96 MiB streamed once -> ~4.3us floor @ 23.3 TB/s.
// CDNA5 paths used: global_load_async_to_lds_b128 + s_wait_asynccnt
// (ASYNCcnt software pipeline, barrier-free inner loop) and ds_add_f32
// LDS histogram atomics. WMMA is inapplicable (no matrix ops in reference).
// ---------------------------------------------------------------------------

#define NSEG   513
#define NBATCH 8
#define TPB    256          // 8 wave32 waves
#define BPB    128          // blocks per batch
#define ITERS  8            // float4 iterations per thread
#define VEC_PER_BATCH (1 << 18)   // 1024*1024/4 float4s per batch image

// --- CDNA5 async global->LDS (per-lane 16B, tracked by ASYNCcnt) -----------
__device__ __forceinline__ void async_ld_b128(unsigned lds_addr, const void* gaddr) {
    asm volatile("global_load_async_to_lds_b128 %0, %1, off"
                 :: "v"(lds_addr), "v"(gaddr) : "memory");
}
#define S_WAIT_ASYNC(n) asm volatile("s_wait_asynccnt " #n ::: "memory")

// --- kernel 1: zero the ws accumulators ------------------------------------
extern "C" __global__ void fb_zero(float* __restrict__ ws, int n) {
    int i = blockIdx.x * blockDim.x + threadIdx.x;
    if (i < n) ws[i] = 0.0f;
}

// --- kernel 2: segmented binning (sum_clean, sum_pred, count per segment) --
extern "C" __global__ __launch_bounds__(TPB)
void fb_bin(const float4* __restrict__ clean, const float4* __restrict__ pred,
            const int4* __restrict__ inst, float* __restrict__ ws)
{
    __shared__ float binC[516];
    __shared__ float binP[516];
    __shared__ float binN[516];
    __shared__ float4 stage[2][3][TPB];   // [buf][array][thread] 16B slots

    const int tid = threadIdx.x;
    const int b   = blockIdx.y;

    for (int i = tid; i < 516; i += TPB) { binC[i] = 0.f; binP[i] = 0.f; binN[i] = 0.f; }
    __syncthreads();

    // contiguous 32KB-per-array region per block: fully coalesced streaming
    const size_t base = (size_t)b * VEC_PER_BATCH
                      + (size_t)blockIdx.x * (TPB * ITERS) + (size_t)tid;

    // prologue: issue iteration 0 into buffer 0
    {
        unsigned lc = (unsigned)(size_t)&stage[0][0][tid];
        unsigned lp = (unsigned)(size_t)&stage[0][1][tid];
        unsigned ln = (unsigned)(size_t)&stage[0][2][tid];
        async_ld_b128(lc, clean + base);
        async_ld_b128(lp, pred  + base);
        async_ld_b128(ln, inst  + base);
    }

    for (int i = 0; i < ITERS; ++i) {
        const int buf = i & 1;
        if (i + 1 < ITERS) {
            const size_t v = base + (size_t)(i + 1) * TPB;
            unsigned lc = (unsigned)(size_t)&stage[buf ^ 1][0][tid];
            unsigned lp = (unsigned)(size_t)&stage[buf ^ 1][1][tid];
            unsigned ln = (unsigned)(size_t)&stage[buf ^ 1][2][tid];
            async_ld_b128(lc, clean + v);
            async_ld_b128(lp, pred  + v);
            async_ld_b128(ln, inst  + v);
            S_WAIT_ASYNC(3);   // 3 next-iter loads may stay in flight; current 3 done
        } else {
            S_WAIT_ASYNC(0);
        }

        // each wave reads only LDS it wrote itself -> no barrier needed
        float4 c = stage[buf][0][tid];
        float4 p = stage[buf][1][tid];
        int4   m = *reinterpret_cast<const int4*>(&stage[buf][2][tid]);

        atomicAdd(&binC[m.x], c.x); atomicAdd(&binP[m.x], p.x); atomicAdd(&binN[m.x], 1.f);
        atomicAdd(&binC[m.y], c.y); atomicAdd(&binP[m.y], p.y); atomicAdd(&binN[m.y], 1.f);
        atomicAdd(&binC[m.z], c.z); atomicAdd(&binP[m.z], p.z); atomicAdd(&binN[m.z], 1.f);
        atomicAdd(&binC[m.w], c.w); atomicAdd(&binP[m.w], p.w); atomicAdd(&binN[m.w], 1.f);
    }
    __syncthreads();

    float* wsC = ws;
    float* wsP = ws + NBATCH * NSEG;
    float* wsN = ws + 2 * NBATCH * NSEG;
    for (int s = tid; s < NSEG; s += TPB) {
        atomicAdd(&wsC[b * NSEG + s], binC[s]);
        atomicAdd(&wsP[b * NSEG + s], binP[s]);
        atomicAdd(&wsN[b * NSEG + s], binN[s]);
    }
}

// --- kernel 3: per-segment logic + reductions -> 4 output scalars ----------
extern "C" __global__ __launch_bounds__(TPB)
void fb_final(const float* __restrict__ ws, float* __restrict__ out)
{
    const float* sumC = ws;
    const float* sumP = ws + NBATCH * NSEG;
    const float* cnt  = ws + 2 * NBATCH * NSEG;

    __shared__ float acc[8];   // sumE,cntE,sumPres,cntPres,sumT,cntT,sumN,cntN
    const int tid = threadIdx.x;

    float sum_lossb = 0.f, sum_validb = 0.f;
    float sum_t = 0.f, sum_ht = 0.f, sum_n = 0.f, sum_hn = 0.f;

    for (int b = 0; b < NBATCH; ++b) {
        if (tid < 8) acc[tid] = 0.f;
        __syncthreads();

        float e_s = 0.f, e_c = 0.f, pr_s = 0.f, pr_c = 0.f;
        float t_s = 0.f, t_c = 0.f, n_s = 0.f, n_c = 0.f;

        for (int s = tid; s < NSEG; s += TPB) {
            float c  = cnt[b * NSEG + s];
            float dn = fmaxf(c, 1.f);
            float cs = sumC[b * NSEG + s] / dn;   // clean score
            float ps = sumP[b * NSEG + s] / dn;   // pred score

            bool valid  = (c >= 8.0f) && (s != 0);
            bool tumor  = valid && (cs >= 0.7f);
            bool normal = valid && (cs <= 0.3f);
            bool pres   = valid && !tumor && !normal;
            bool enh    = tumor || normal;

            float tgt = tumor  ? fminf(cs + 0.08f, 1.f)
                      : normal ? fmaxf(cs - 0.08f, 0.f)
                               : cs;
            float d  = ps - tgt;
            float ad = fabsf(d);
            float sl = (ad < 1.f) ? 0.5f * d * d : ad - 0.5f;

            if (enh)  { e_s  += sl; e_c  += 1.f; }
            if (pres) { pr_s += sl; pr_c += 1.f; }
            bool tm = valid && (ps > 0.5f);
            bool nm = valid && !(ps > 0.5f);
            if (tm) { t_s += ps; t_c += 1.f; }
            if (nm) { n_s += ps; n_c += 1.f; }
        }
        atomicAdd(&acc[0], e_s);  atomicAdd(&acc[1], e_c);
        atomicAdd(&acc[2], pr_s); atomicAdd(&acc[3], pr_c);
        atomicAdd(&acc[4], t_s);  atomicAdd(&acc[5], t_c);
        atomicAdd(&acc[6], n_s);  atomicAdd(&acc[7], n_c);
        __syncthreads();

        if (tid == 0) {
            float cntE = acc[1], cntP = acc[3];
            float loss_enh  = acc[0] / fmaxf(cntE, 1.f);
            float loss_pres = acc[2] / fmaxf(cntP, 1.f);
            float hasE = (cntE > 0.f) ? 1.f : 0.f;
            float hasP = (cntP > 0.f) ? 1.f : 0.f;
            float c2   = hasE + hasP;
            float lossb  = (loss_enh * hasE + 0.5f * loss_pres * hasP) / fmaxf(c2, 1.f);
            float validb = (c2 > 0.f) ? 1.f : 0.f;
            sum_lossb  += lossb * validb;
            sum_validb += validb;

            float hT = (acc[5] > 0.f) ? 1.f : 0.f;
            float hN = (acc[7] > 0.f) ? 1.f : 0.f;
            sum_t  += (acc[4] / fmaxf(acc[5], 1.f)) * hT;  sum_ht += hT;
            sum_n  += (acc[6] / fmaxf(acc[7], 1.f)) * hN;  sum_hn += hN;
        }
        __syncthreads();   // protect acc[] re-zero for next batch
    }

    if (tid == 0) {
        float loss_prob = sum_lossb / fmaxf(sum_validb, 1.f);
        float avg_t = (sum_ht > 0.f) ? (sum_t / fmaxf(sum_ht, 1.f)) : -1.f;
        float avg_n = (sum_hn > 0.f) ? (sum_n / fmaxf(sum_hn, 1.f)) : -1.f;
        bool anyv = (sum_validb > 0.f);
        out[0] = anyv ? loss_prob : 0.f;
        out[1] = 0.f;
        out[2] = anyv ? avg_t : -1.f;
        out[3] = anyv ? avg_n : -1.f;
    }
}

// ---------------------------------------------------------------------------
extern "C" void kernel_launch(void* const* d_in, const int* in_sizes, int n_in,
                              void* d_out, int out_size, void* d_ws, size_t ws_size,
                              hipStream_t stream) {
    const float4* clean = (const float4*)d_in[0];
    const float4* pred  = (const float4*)d_in[1];
    const int4*   inst  = (const int4*)d_in[2];
    float* ws  = (float*)d_ws;
    float* out = (float*)d_out;

    const int nacc = 3 * NBATCH * NSEG;   // 12312 floats
    fb_zero<<<(nacc + TPB - 1) / TPB, TPB, 0, stream>>>(ws, nacc);

    dim3 grid(BPB, NBATCH);               // 128 x 8 blocks, 256 threads each
    fb_bin<<<grid, TPB, 0, stream>>>(clean, pred, inst, ws);

    fb_final<<<1, TPB, 0, stream>>>(ws, out);
}